// OneHopSumNodeLabelAggregator_2568390443270
// MI455X (gfx1250) — compile-verified
//
#include <hip/hip_runtime.h>
#include <cstdint>
#include <cstddef>

// ---------------------------------------------------------------------------
// OneHopSumNodeLabelAggregator for MI455X (gfx1250)
//
// out[b, 0:128]   = x[b]                     (async global->LDS->global DMA)
// out[b, 128:256] = sum_{e: dst[e]==b} x[src[e]]
//
// Scatter-sum is done atomic-free via an on-device CSR build:
//   hist(dst) -> exclusive scan -> ticketed bucket scatter of src ids ->
//   wave-per-row register-accumulated gather (pure coalesced b128 traffic).
// 82M f32 L2 atomics (the bottleneck of the naive version) become
// 1.3M int atomics + bandwidth-floor reads/writes. x (51 MB) and the
// bucket array (2.6 MB) live in the 192 MB L2.
// ---------------------------------------------------------------------------

#define FEAT      128
#define THREADS   256

// ---------------- Kernel 1: copy x[:B] into out[:, :128] (async DMA path) ---
__global__ void __launch_bounds__(THREADS)
copy_kernel(const float* __restrict__ x,
            float* __restrict__ out,
            int total_f4,
            int zero_second_half)
{
    __shared__ __align__(16) float lds_buf[THREADS * 4];

    int idx = blockIdx.x * THREADS + threadIdx.x;   // one float4 per lane
    if (idx >= total_f4) return;

    int r = idx >> 5;        // output row
    int c = idx & 31;        // float4 column (0..31)

    // LDS byte offset of this lane's private 16B staging slot (low 32 bits of
    // the generic address of a __shared__ object are the LDS byte offset).
    unsigned lds_off = (unsigned)(size_t)(&lds_buf[threadIdx.x * 4]);
    unsigned src_off = (unsigned)idx * 16u;             // byte off into x
    unsigned dst_off = (unsigned)(r * 64 + c) * 16u;    // byte off into out

    asm volatile("global_load_async_to_lds_b128 %0, %1, %2"
                 :: "v"(lds_off), "v"(src_off), "s"(x) : "memory");
    asm volatile("s_wait_asynccnt 0" ::: "memory");
    asm volatile("global_store_async_from_lds_b128 %0, %1, %2"
                 :: "v"(dst_off), "v"(lds_off), "s"(out) : "memory");

    if (zero_second_half) {      // only needed by the atomic fallback path
        float4 z = make_float4(0.f, 0.f, 0.f, 0.f);
        ((float4*)out)[(size_t)r * 64 + 32 + c] = z;
    }
}

// ---------------- CSR build ------------------------------------------------
__global__ void __launch_bounds__(THREADS)
zero_ints_kernel(int* __restrict__ p, int n)
{
    int i = blockIdx.x * THREADS + threadIdx.x;
    if (i < n) p[i] = 0;
}

__global__ void __launch_bounds__(THREADS)
hist_kernel(const int* __restrict__ dst, int* __restrict__ deg,
            int n_edges, int batch)
{
    int e = blockIdx.x * THREADS + threadIdx.x;
    if (e >= n_edges) return;
    int d = dst[e];
    if (d < batch) atomicAdd(&deg[d], 1);
}

__device__ __forceinline__ int wave_incl_scan(int v, int lane)
{
    #pragma unroll
    for (int d = 1; d < 32; d <<= 1) {
        int t = __shfl_up(v, d, 32);
        if (lane >= d) v += t;
    }
    return v;
}

// Single-block exclusive scan of deg[0..B-1] -> row_start[0..B] (+ cursor copy).
__global__ void __launch_bounds__(1024)
scan_kernel(const int* __restrict__ deg,
            int* __restrict__ row_start,
            int* __restrict__ cursor,
            int B)
{
    __shared__ int wsum[32];
    __shared__ int woff[32];
    __shared__ int stotal;

    int lane = threadIdx.x & 31;
    int wid  = threadIdx.x >> 5;
    int running = 0;
    int nChunks = (B + 1023) >> 10;

    for (int c = 0; c < nChunks; ++c) {
        int i = (c << 10) + threadIdx.x;
        int v = (i < B) ? deg[i] : 0;
        int incl = wave_incl_scan(v, lane);          // intra-wave, no barriers
        if (lane == 31) wsum[wid] = incl;
        __syncthreads();
        if (wid == 0) {
            int w  = wsum[lane];
            int wi = wave_incl_scan(w, lane);
            woff[lane] = wi - w;                     // exclusive wave offsets
            if (lane == 31) stotal = wi;             // chunk total
        }
        __syncthreads();
        int excl = (incl - v) + woff[wid] + running;
        if (i < B) { row_start[i] = excl; cursor[i] = excl; }
        running += stotal;
        __syncthreads();                             // protect wsum/stotal
    }
    if (threadIdx.x == 0) row_start[B] = running;
}

__global__ void __launch_bounds__(THREADS)
scatter_src_kernel(const int* __restrict__ src, const int* __restrict__ dst,
                   int* __restrict__ cursor, int* __restrict__ srcs,
                   int n_edges, int batch)
{
    int e = blockIdx.x * THREADS + threadIdx.x;
    if (e >= n_edges) return;
    int d = dst[e];
    if (d < batch) {
        int pos = atomicAdd(&cursor[d], 1);          // cheap int ticket
        srcs[pos] = src[e];
    }
}

// ---------------- Gather: wave32 per output row, atomic-free ---------------
__global__ void __launch_bounds__(THREADS)
gather_rows_kernel(const float* __restrict__ x,
                   const int* __restrict__ row_start,
                   const int* __restrict__ srcs,
                   float* __restrict__ out,
                   int B)
{
    int row  = (int)((blockIdx.x * THREADS + threadIdx.x) >> 5);
    int lane = threadIdx.x & 31;
    if (row >= B) return;

    int beg = row_start[row];
    int end = row_start[row + 1];

    float4 acc = make_float4(0.f, 0.f, 0.f, 0.f);
    for (int j = beg; j < end; ++j) {
        int s = srcs[j];                              // wave-uniform address
        const float4* r4 = (const float4*)(x + (size_t)s * FEAT);
        float4 v = r4[lane];                          // coalesced b128, L2-hot
        acc.x += v.x; acc.y += v.y; acc.z += v.z; acc.w += v.w;
    }
    ((float4*)out)[(size_t)row * 64 + 32 + lane] = acc;  // single b128 store
}

// ---------------- Fallback: direct f32 atomic scatter (if ws too small) ----
__global__ void __launch_bounds__(THREADS)
scatter_add_kernel(const float* __restrict__ x,
                   const int* __restrict__ src,
                   const int* __restrict__ dst,
                   float* __restrict__ out,
                   int n_edges, int batch)
{
    int wave = (int)((blockIdx.x * THREADS + threadIdx.x) >> 5);
    int lane = threadIdx.x & 31;
    if (wave >= n_edges) return;
    int d = dst[wave];
    if (d >= batch) return;
    int s = src[wave];
    const float4* row = (const float4*)(x + (size_t)s * FEAT);
    float4 v = row[lane];
    float* o = out + (size_t)d * (2 * FEAT) + FEAT + (size_t)lane * 4;
    atomicAdd(o + 0, v.x);
    atomicAdd(o + 1, v.y);
    atomicAdd(o + 2, v.z);
    atomicAdd(o + 3, v.w);
}

// ---------------------------------------------------------------------------
extern "C" void kernel_launch(void* const* d_in, const int* in_sizes, int n_in,
                              void* d_out, int out_size, void* d_ws, size_t ws_size,
                              hipStream_t stream)
{
    const float* x  = (const float*)d_in[0];
    const int*   ei = (const int*)d_in[1];
    int batch   = out_size / (2 * FEAT);      // 50000 (host-known, capture-safe)
    int n_edges = in_sizes[1] / 2;            // 640000
    const int* src = ei;
    const int* dst = ei + n_edges;
    float* out = (float*)d_out;

    int total_f4 = batch * 32;
    int blocks_copy = (total_f4 + THREADS - 1) / THREADS;
    int blocks_edge = (n_edges + THREADS - 1) / THREADS;

    // Workspace: deg[B+1] | row_start[B+1] | cursor[B+1] | srcs[E]
    size_t need_ints = (size_t)3 * (batch + 1) + (size_t)n_edges;
    bool use_csr = ws_size >= need_ints * sizeof(int);

    if (use_csr) {
        int* deg       = (int*)d_ws;
        int* row_start = deg + (batch + 1);
        int* cursor    = row_start + (batch + 1);
        int* srcs      = cursor + (batch + 1);

        copy_kernel<<<blocks_copy, THREADS, 0, stream>>>(x, out, total_f4, 0);

        int nclr = batch + 1;
        zero_ints_kernel<<<(nclr + THREADS - 1) / THREADS, THREADS, 0, stream>>>(deg, nclr);
        hist_kernel<<<blocks_edge, THREADS, 0, stream>>>(dst, deg, n_edges, batch);
        scan_kernel<<<1, 1024, 0, stream>>>(deg, row_start, cursor, batch);
        scatter_src_kernel<<<blocks_edge, THREADS, 0, stream>>>(src, dst, cursor,
                                                                srcs, n_edges, batch);
        int blocks_gather = (batch * 32 + THREADS - 1) / THREADS;  // wave per row
        gather_rows_kernel<<<blocks_gather, THREADS, 0, stream>>>(x, row_start,
                                                                  srcs, out, batch);
    } else {
        // Fallback: zero second half, then direct f32 atomic scatter.
        copy_kernel<<<blocks_copy, THREADS, 0, stream>>>(x, out, total_f4, 1);
        int waves_per_block = THREADS / 32;
        int blocks2 = (n_edges + waves_per_block - 1) / waves_per_block;
        scatter_add_kernel<<<blocks2, THREADS, 0, stream>>>(x, src, dst, out,
                                                            n_edges, batch);
    }
}